// Custom_SiLU_75393855914398
// MI455X (gfx1250) — compile-verified
//
#include <hip/hip_runtime.h>

#define SEQ_LEN   8192
#define HIDDEN    4096
#define THREADS   256
#define CHUNK_ELTS (THREADS * 4)        // 1024 int32 per chunk
#define CHUNK_BYTES (CHUNK_ELTS * 4)    // 4096 bytes per chunk
#define NCHUNKS   (HIDDEN / CHUNK_ELTS) // 4 chunks per row

typedef int   v4i __attribute__((ext_vector_type(4)));
typedef float v4f __attribute__((ext_vector_type(4)));

// ---- CDNA5 async global->LDS copy (ASYNCcnt-tracked, no VGPR held in flight) ----
// GVS mode: LDS address = LDS base + vdst VGPR offset; global address = 64-bit saddr + 32-bit voffset
__device__ __forceinline__ void async_load_b128_to_lds(unsigned lds_byte_off,
                                                       unsigned global_byte_off,
                                                       const int* saddr) {
    asm volatile("global_load_async_to_lds_b128 %0, %1, %2"
                 :
                 : "v"(lds_byte_off), "v"(global_byte_off), "s"(saddr)
                 : "memory");
}

__device__ __forceinline__ unsigned lds_offset_of(const void* p) {
    // For HIP shared pointers, the low 32 bits of the flat address are the
    // DS-relative byte address.
    return (unsigned)(unsigned long long)p;
}

__device__ __forceinline__ float fast_sigmoid(float xf) {
    // sigmoid(x) = 1 / (1 + exp(-x)); exp(-x) = exp2(-x * log2(e)) -> v_exp_f32
    float e = __builtin_amdgcn_exp2f(xf * -1.44269504088896340736f);
    return __builtin_amdgcn_rcpf(1.0f + e);
}

__device__ __forceinline__ float quantize1(float v, float inv_s) {
    float q = __builtin_rintf(v * inv_s);     // v_rndne_f32 (round-to-nearest-even)
    return fminf(fmaxf(q, -127.0f), 127.0f);  // clip to int8 range
}

__global__ __launch_bounds__(THREADS) void
silu_quant_kernel(const int* __restrict__ x,
                  const float* __restrict__ scale_x,
                  const float* __restrict__ scale_y,
                  const float* __restrict__ scale_out,
                  float* __restrict__ out) {
    // 16 KB: one LDS buffer per chunk -> all 4 async loads in flight at once.
    __shared__ __align__(16) int lds[NCHUNKS * CHUNK_ELTS];

    const int row = blockIdx.x;          // one block == one row of 4096 elements
    const int tid = threadIdx.x;

    const int*   xrow = x   + (size_t)row * HIDDEN;
    float*       orow = out + (size_t)row * HIDDEN;

    const unsigned lane_bytes = (unsigned)tid * 16u;               // 16 B per lane
    const unsigned lds_base   = lds_offset_of(&lds[0]) + lane_bytes;

    // Issue ALL row chunks immediately: 16 KB of per-block in-flight HBM traffic
    // tracked by ASYNCcnt, holding zero VGPRs while pending.
#pragma unroll
    for (int c = 0; c < NCHUNKS; ++c) {
        async_load_b128_to_lds(lds_base + (unsigned)c * CHUNK_BYTES,
                               lane_bytes + (unsigned)c * CHUNK_BYTES,
                               xrow);
    }

    // Per-row scales: uniform across the block -> scalar loads / v_s_rcp_f32,
    // fetched while the async copies stream.
    const float s_x = scale_x[row];
    const float s_y = scale_y[row];
    const float s_o = scale_out[row];
    const float inv_sy = __builtin_amdgcn_rcpf(s_y);
    const float inv_so = __builtin_amdgcn_rcpf(s_o);

#pragma unroll
    for (int c = 0; c < NCHUNKS; ++c) {
        // Async loads complete in order: waiting ASYNCcnt <= (NCHUNKS-1-c)
        // guarantees chunk c has landed in LDS.
        if      (c == 0) asm volatile("s_wait_asynccnt 0x3" ::: "memory");
        else if (c == 1) asm volatile("s_wait_asynccnt 0x2" ::: "memory");
        else if (c == 2) asm volatile("s_wait_asynccnt 0x1" ::: "memory");
        else             asm volatile("s_wait_asynccnt 0x0" ::: "memory");

        // Each wave reads back exactly the LDS bytes its own lanes wrote:
        // no workgroup barrier required, ASYNCcnt orders producer->consumer.
        v4i xi = *(const v4i*)(&lds[c * CHUNK_ELTS + tid * 4]);

        v4f o;
#pragma unroll
        for (int k = 0; k < 4; ++k) {
            float xf  = (float)xi[k] * s_x;                 // dequant x
            float sig = fast_sigmoid(xf);                   // sigmoid
            float yq  = quantize1(sig, inv_sy);             // requant sigmoid
            float yf  = yq * s_y;                           // dequant
            float of  = xf * yf;                            // SiLU product
            o[k]      = quantize1(of, inv_so);              // final int8-valued requant
        }

        // Streaming store: non-temporal so 128 MB of stores doesn't thrash L2
        __builtin_nontemporal_store(o, (v4f*)(orow + c * CHUNK_ELTS + tid * 4));
    }

    // Tuple output tail: scale_out passthrough (one float per row)
    if (tid == 0) {
        out[(size_t)SEQ_LEN * HIDDEN + row] = s_o;
    }
}

extern "C" void kernel_launch(void* const* d_in, const int* in_sizes, int n_in,
                              void* d_out, int out_size, void* d_ws, size_t ws_size,
                              hipStream_t stream) {
    const int*   x         = (const int*)d_in[0];
    const float* scale_x   = (const float*)d_in[1];
    const float* scale_y   = (const float*)d_in[2];
    const float* scale_out = (const float*)d_in[3];
    float*       out       = (float*)d_out;

    silu_quant_kernel<<<SEQ_LEN, THREADS, 0, stream>>>(x, scale_x, scale_y,
                                                       scale_out, out);
}